// fromNorm_mix_20358144983646
// MI455X (gfx1250) — compile-verified
//
#include <hip/hip_runtime.h>
#include <hip/hip_bf16.h>

// Problem constants (from reference)
#define Bn 8192
#define En 16
#define Zn 256
#define Hn 1024
#define Un 256
#define LDS_STRIDE 260   // 256 + 4: rows land on distinct banks (4*r mod 64 distinct for r<16)

typedef __attribute__((ext_vector_type(2))) float v2f;
typedef __attribute__((ext_vector_type(8))) float v8f;

// ---------------- prologue: group samples by expert ----------------

__global__ void zero_ws_kernel(int* wsI) {
    // zero counts[16], offsets[16], cursor[16], pad[16]
    if (threadIdx.x < 64) wsI[threadIdx.x] = 0;
}

__global__ void count_experts_kernel(const int* __restrict__ rng, int* __restrict__ counts) {
    int i = blockIdx.x * blockDim.x + threadIdx.x;
    if (i < Bn) atomicAdd(&counts[rng[i]], 1);
}

__global__ void scan_offsets_kernel(const int* __restrict__ counts, int* __restrict__ offsets) {
    if (threadIdx.x == 0) {
        int acc = 0;
        for (int e = 0; e < En; ++e) { offsets[e] = acc; acc += counts[e]; }
    }
}

__global__ void scatter_perm_kernel(const int* __restrict__ rng,
                                    const int* __restrict__ offsets,
                                    int* __restrict__ cursor,
                                    int* __restrict__ perm) {
    int i = blockIdx.x * blockDim.x + threadIdx.x;
    if (i < Bn) {
        int e = rng[i];
        int pos = atomicAdd(&cursor[e], 1);
        perm[offsets[e] + pos] = i;
    }
}

// ---------------- main grouped-GEMM kernel (f32 WMMA) ----------------

__global__ __launch_bounds__(256) void moe_wmma_kernel(
    const float* __restrict__ z,   const float* __restrict__ u,
    const float* __restrict__ Wz,  const float* __restrict__ bz,
    const float* __restrict__ Wh,  const float* __restrict__ bh,
    const float* __restrict__ Wu,
    const int* __restrict__ counts, const int* __restrict__ offsets,
    const int* __restrict__ perm,
    float* __restrict__ out_hh, float* __restrict__ out_v)
{
    __shared__ float zs[16 * LDS_STRIDE];
    __shared__ float as[16 * LDS_STRIDE];
    __shared__ float us[16 * LDS_STRIDE];
    __shared__ int   ssam[16];
    __shared__ int   sval[16];

    const int e   = blockIdx.y;
    const int t   = blockIdx.x;
    const int cnt = counts[e];
    if (t * 16 >= cnt) return;          // uniform per-block exit
    const int start = offsets[e];
    const int tid   = threadIdx.x;

    if (tid < 16) {
        int p = t * 16 + tid;
        int q = (p < cnt) ? p : (cnt - 1);   // clamp padding rows to a real sample
        ssam[tid] = perm[start + q];
        sval[tid] = (p < cnt) ? 1 : 0;
    }
    __syncthreads();

    // Gather the 16 sample rows of z and u into LDS (coalesced: 1 row / iter).
    #pragma unroll 4
    for (int it = 0; it < 16; ++it) {
        int s = ssam[it];
        zs[it * LDS_STRIDE + tid] = z[(size_t)s * Zn + tid];
        us[it * LDS_STRIDE + tid] = u[(size_t)s * Un + tid];
    }
    __syncthreads();

    const int lane = tid & 31;
    const int wave = tid >> 5;     // 8 waves
    const int ln   = lane & 15;    // N (B/C/D) or M (A) index of this lane
    const int hi   = lane >> 4;    // selects K pair (A/B) and M half (C/D)

    // ---------- Phase A: a = z @ Wz[e]^T + bz[e]  (16 x 256) -> LDS ----------
    const float* Wze = Wz + (size_t)e * Zn * Zn;
    for (int tt = 0; tt < 2; ++tt) {
        const int nb = (wave * 2 + tt) * 16;
        const int n  = nb + ln;
        const float* wrow = Wze + (size_t)n * Zn;
        const float* arow = zs + ln * LDS_STRIDE + 2 * hi;
        v8f acc = {};
        #pragma unroll 8
        for (int k = 0; k < Zn; k += 4) {
            v2f a = *(const v2f*)(arow + k);
            v2f b = *(const v2f*)(wrow + k + 2 * hi);
            acc = __builtin_amdgcn_wmma_f32_16x16x4_f32(
                false, a, false, b, (short)0, acc, false, false);
        }
        const float bias = bz[e * Zn + n];
        #pragma unroll
        for (int r = 0; r < 8; ++r)
            as[(r + 8 * hi) * LDS_STRIDE + n] = acc[r] + bias;
    }
    __syncthreads();

    // ---------- Phase H: h = relu(a @ Wh[e]^T + bh[e])  (16 x 1024) ----------
    const float* Whe = Wh + (size_t)e * Hn * Zn;
    for (int tt = 0; tt < 8; ++tt) {
        const int nb = (wave * 8 + tt) * 16;
        const int n  = nb + ln;
        const float* wrow = Whe + (size_t)n * Zn;
        const float* arow = as + ln * LDS_STRIDE + 2 * hi;
        v8f acc = {};
        #pragma unroll 8
        for (int k = 0; k < Zn; k += 4) {
            v2f a = *(const v2f*)(arow + k);
            v2f b = *(const v2f*)(wrow + k + 2 * hi);
            acc = __builtin_amdgcn_wmma_f32_16x16x4_f32(
                false, a, false, b, (short)0, acc, false, false);
        }
        const float bias = bh[e * Hn + n];
        #pragma unroll
        for (int r = 0; r < 8; ++r) {
            int m = r + 8 * hi;
            if (sval[m])
                out_hh[(size_t)ssam[m] * Hn + n] = fmaxf(acc[r] + bias, 0.0f);
        }
    }

    // ---------- Phase V: v = u @ Wu[e]^T  (16 x 256) ----------
    const float* Wue = Wu + (size_t)e * Un * Un;
    for (int tt = 0; tt < 2; ++tt) {
        const int nb = (wave * 2 + tt) * 16;
        const int n  = nb + ln;
        const float* wrow = Wue + (size_t)n * Un;
        const float* arow = us + ln * LDS_STRIDE + 2 * hi;
        v8f acc = {};
        #pragma unroll 8
        for (int k = 0; k < Un; k += 4) {
            v2f a = *(const v2f*)(arow + k);
            v2f b = *(const v2f*)(wrow + k + 2 * hi);
            acc = __builtin_amdgcn_wmma_f32_16x16x4_f32(
                false, a, false, b, (short)0, acc, false, false);
        }
        #pragma unroll
        for (int r = 0; r < 8; ++r) {
            int m = r + 8 * hi;
            if (sval[m])
                out_v[(size_t)ssam[m] * Un + n] = acc[r];
        }
    }
}

// ---------------- launch ----------------

extern "C" void kernel_launch(void* const* d_in, const int* in_sizes, int n_in,
                              void* d_out, int out_size, void* d_ws, size_t ws_size,
                              hipStream_t stream) {
    const float* z   = (const float*)d_in[0];
    const float* u   = (const float*)d_in[1];
    const int*   rng = (const int*)d_in[2];
    const float* Wz  = (const float*)d_in[3];
    const float* bz  = (const float*)d_in[4];
    const float* Wh  = (const float*)d_in[5];
    const float* bh  = (const float*)d_in[6];
    const float* Wu  = (const float*)d_in[7];

    float* out    = (float*)d_out;
    float* out_hh = out;
    float* out_v  = out + (size_t)Bn * Hn;

    int* wsI     = (int*)d_ws;
    int* counts  = wsI;
    int* offsets = wsI + 16;
    int* cursor  = wsI + 32;
    int* perm    = wsI + 64;

    zero_ws_kernel<<<1, 64, 0, stream>>>(wsI);
    count_experts_kernel<<<Bn / 256, 256, 0, stream>>>(rng, counts);
    scan_offsets_kernel<<<1, 32, 0, stream>>>(counts, offsets);
    scatter_perm_kernel<<<Bn / 256, 256, 0, stream>>>(rng, offsets, cursor, perm);

    dim3 grid(Bn / 16, En);
    moe_wmma_kernel<<<grid, 256, 0, stream>>>(z, u, Wz, bz, Wh, bh, Wu,
                                              counts, offsets, perm, out_hh, out_v);
}